// ChamferDistance_755914244601
// MI455X (gfx1250) — compile-verified
//
#include <hip/hip_runtime.h>

// CDNA5 (gfx1250) wave32 WMMA types
typedef __attribute__((ext_vector_type(2))) float v2f;
typedef __attribute__((ext_vector_type(8))) float v8f;

#define TILE 16
#define QTILES 2                       // 16-row query strips per wave
#define WAVES_PER_BLOCK 8
#define BLOCK_THREADS (WAVES_PER_BLOCK * 32)
#define BIGF 3.402823466e+38f          // FLT_MAX

// Raw CDNA5 min ops: bypass strict-IEEE fminf lowering, which inserts a
// v_max_num_f32 x,x,x sNaN-canonicalize per operand.  Squared distances are
// NaN-free, so bare v_min_num / v_min3_num are exact and half the cost.
__device__ __forceinline__ float fmin_asm(float a, float b) {
    float r;
    asm("v_min_num_f32 %0, %1, %2" : "=v"(r) : "v"(a), "v"(b));
    return r;
}
__device__ __forceinline__ float fmin3_asm(float a, float b, float c) {
    float r;
    asm("v_min3_num_f32 %0, %1, %2, %3" : "=v"(r) : "v"(a), "v"(b), "v"(c));
    return r;
}

// Precompute ||p||^2 per point into workspace.
__global__ void chamfer_sqnorm_kernel(const float* __restrict__ pts,
                                      float* __restrict__ out, int count) {
    int i = blockIdx.x * blockDim.x + threadIdx.x;
    if (i < count) {
        float x = pts[3 * i + 0];
        float y = pts[3 * i + 1];
        float z = pts[3 * i + 2];
        out[i] = x * x + y * y + z * z;
    }
}

// One wave owns QTILES 16-row strips of "query" points and sweeps all
// "target" points two 16-wide tiles at a time.  The K=4 WMMA carries the
// squared norm in the spare 4th K slot:
//   A row = (-2qx, -2qy, -2qz, 1),  B col = (tx, ty, tz, ||t||^2)
//   D = A*B + 0 = ||t||^2 - 2 q.t          (C is inline zero)
// Both D tiles fold into the running row-min with one v_min3_num_f32 per
// element.  shfl_xor(1,2,4,8) reduces over the 16 columns inside each
// half-wave (C/D layout: lanes 0-15 rows 0-7, lanes 16-31 rows 8-15);
// ||q||^2 is added once at the store.  Called twice with roles swapped
// -> d1, d2.  No atomics, deterministic.
template <bool PRECOMP_SQ>
__global__ __launch_bounds__(BLOCK_THREADS)
void chamfer_min_wmma_kernel(const float* __restrict__ qry,   // [B, Nq, 3]
                             const float* __restrict__ qsq,   // [B*Nq] or null
                             const float* __restrict__ tgt,   // [B, Nt, 3]
                             const float* __restrict__ tsq,   // [B*Nt] or null
                             float* __restrict__ out,         // [B*Nq]
                             int Nq, int Nt, int totalGroups)
{
    const int lane  = threadIdx.x & 31;
    const int wave  = threadIdx.x >> 5;
    const int group = blockIdx.x * WAVES_PER_BLOCK + wave;
    if (group >= totalGroups) return;   // uniform per wave

    const int groupsPerBatch = Nq / (TILE * QTILES);
    const int b     = group / groupsPerBatch;
    const int nBase = group * (TILE * QTILES);  // flat row index over B*Nq
    const int col   = lane & 15;
    const bool hi   = (lane >= 16);

    // ---- A fragments: 16x4 f32 per strip, row = lane&15 ----
    v2f afrag[QTILES];
    #pragma unroll
    for (int qt = 0; qt < QTILES; ++qt) {
        const int n = nBase + qt * TILE + col;
        const float x = qry[3 * n + 0];
        const float y = qry[3 * n + 1];
        const float z = qry[3 * n + 2];
        afrag[qt].x = -2.0f * (hi ? z : x);       // K=0 (lo) / K=2 (hi)
        afrag[qt].y = hi ? 1.0f : (-2.0f * y);    // K=1 (lo) / K=3 = 1 (hi)
    }

    const float* tb = tgt + (size_t)b * Nt * 3;
    const float* tq = PRECOMP_SQ ? (tsq + (size_t)b * Nt) : nullptr;

    v8f acc[QTILES];
    #pragma unroll
    for (int qt = 0; qt < QTILES; ++qt)
        acc[qt] = (v8f){ BIGF, BIGF, BIGF, BIGF, BIGF, BIGF, BIGF, BIGF };

    const v8f czero = {};   // constant-zero C -> inline 0 operand

    // Helper lambda: build a B fragment (+||t||^2 in K=3) for column mBase+col
    auto loadB = [&](int mBase) -> v2f {
        const int m = mBase + col;
        const float tx = tb[3 * m + 0];
        const float ty = tb[3 * m + 1];
        const float tz = tb[3 * m + 2];
        float sq;
        if (PRECOMP_SQ) {
            sq = tq[m];
        } else {
            float p = hi ? tz * tz : (tx * tx + ty * ty);
            sq = p + __shfl_xor(p, 16, 32);
        }
        v2f bf;
        bf.x = hi ? tz : tx;                      // K=0 (lo) / K=2 (hi)
        bf.y = hi ? sq : ty;                      // K=1 (lo) / K=3 = ||t||^2
        return bf;
    };

    // ---- main sweep: two 16-col target tiles per iteration ----
    const int NtMain = Nt & ~(2 * TILE - 1);
    for (int mBase = 0; mBase < NtMain; mBase += 2 * TILE) {
        const v2f bf0 = loadB(mBase);
        const v2f bf1 = loadB(mBase + TILE);

        #pragma unroll
        for (int qt = 0; qt < QTILES; ++qt) {
            v8f d0 = __builtin_amdgcn_wmma_f32_16x16x4_f32(
                         false, afrag[qt], false, bf0, (short)0, czero,
                         false, false);
            v8f d1 = __builtin_amdgcn_wmma_f32_16x16x4_f32(
                         false, afrag[qt], false, bf1, (short)0, czero,
                         false, false);
            #pragma unroll
            for (int r = 0; r < 8; ++r)
                acc[qt][r] = fmin3_asm(acc[qt][r], d0[r], d1[r]);
        }
    }
    // ---- tail (Nt % 32): single 16-col tile ----
    for (int mBase = NtMain; mBase < Nt; mBase += TILE) {
        const v2f bf0 = loadB(mBase);
        #pragma unroll
        for (int qt = 0; qt < QTILES; ++qt) {
            v8f d0 = __builtin_amdgcn_wmma_f32_16x16x4_f32(
                         false, afrag[qt], false, bf0, (short)0, czero,
                         false, false);
            #pragma unroll
            for (int r = 0; r < 8; ++r)
                acc[qt][r] = fmin_asm(acc[qt][r], d0[r]);
        }
    }

    // ---- reduce over the 16 columns inside each 16-lane half ----
    #pragma unroll
    for (int off = 1; off < 16; off <<= 1) {
        #pragma unroll
        for (int qt = 0; qt < QTILES; ++qt)
            #pragma unroll
            for (int r = 0; r < 8; ++r)
                acc[qt][r] = fmin_asm(acc[qt][r], __shfl_xor(acc[qt][r], off, 32));
    }

    // ---- lanes 0 / 16 store rows 0..7 / 8..15, adding ||q||^2 ----
    if (col == 0) {
        const int rowOff = hi ? 8 : 0;
        #pragma unroll
        for (int qt = 0; qt < QTILES; ++qt) {
            #pragma unroll
            for (int r = 0; r < 8; ++r) {
                const int n = nBase + qt * TILE + rowOff + r;
                float q2;
                if (PRECOMP_SQ) {
                    q2 = qsq[n];
                } else {
                    const float x = qry[3 * n + 0];
                    const float y = qry[3 * n + 1];
                    const float z = qry[3 * n + 2];
                    q2 = x * x + y * y + z * z;
                }
                out[n] = acc[qt][r] + q2;
            }
        }
    }
}

extern "C" void kernel_launch(void* const* d_in, const int* in_sizes, int n_in,
                              void* d_out, int out_size, void* d_ws, size_t ws_size,
                              hipStream_t stream) {
    const float* xyz1 = (const float*)d_in[0];   // [B, N, 3] f32
    const float* xyz2 = (const float*)d_in[1];   // [B, M, 3] f32

    const int B = 4;
    const int N = in_sizes[0] / (B * 3);
    const int M = in_sizes[1] / (B * 3);

    float* d1 = (float*)d_out;                   // [B, N]
    float* d2 = d1 + (size_t)B * N;              // [B, M]

    const size_t sqBytes = (size_t)B * (N + M) * sizeof(float);
    const bool usePre = (d_ws != nullptr) && (ws_size >= sqBytes);

    float* sq1 = (float*)d_ws;                   // [B*N]
    float* sq2 = sq1 + (size_t)B * N;            // [B*M]

    if (usePre) {
        const int c1 = B * N, c2 = B * M;
        chamfer_sqnorm_kernel<<<(c1 + 255) / 256, 256, 0, stream>>>(xyz1, sq1, c1);
        chamfer_sqnorm_kernel<<<(c2 + 255) / 256, 256, 0, stream>>>(xyz2, sq2, c2);
    }

    const int rowsPerWave = TILE * QTILES;
    // d1: queries = xyz1, targets = xyz2
    {
        const int groups = (B * N) / rowsPerWave;
        const int grid = (groups + WAVES_PER_BLOCK - 1) / WAVES_PER_BLOCK;
        if (usePre)
            chamfer_min_wmma_kernel<true><<<grid, BLOCK_THREADS, 0, stream>>>(
                xyz1, sq1, xyz2, sq2, d1, N, M, groups);
        else
            chamfer_min_wmma_kernel<false><<<grid, BLOCK_THREADS, 0, stream>>>(
                xyz1, nullptr, xyz2, nullptr, d1, N, M, groups);
    }
    // d2: queries = xyz2, targets = xyz1
    {
        const int groups = (B * M) / rowsPerWave;
        const int grid = (groups + WAVES_PER_BLOCK - 1) / WAVES_PER_BLOCK;
        if (usePre)
            chamfer_min_wmma_kernel<true><<<grid, BLOCK_THREADS, 0, stream>>>(
                xyz2, sq2, xyz1, sq1, d2, M, N, groups);
        else
            chamfer_min_wmma_kernel<false><<<grid, BLOCK_THREADS, 0, stream>>>(
                xyz2, nullptr, xyz1, nullptr, d2, M, N, groups);
    }
}